// Att2in2Core_43465069035697
// MI455X (gfx1250) — compile-verified
//
#include <hip/hip_runtime.h>
#include <hip/hip_bf16.h>
#include <math.h>

#define RNN   1024
#define ATTH  512
#define ENCSZ 1024
#define BSZ   256
#define LSZ   196
#define RSZ   100
#define MIN_VALUE -100000000.0f

typedef __attribute__((ext_vector_type(2))) float v2f;
typedef __attribute__((ext_vector_type(8))) float v8f;

__device__ __forceinline__ float wave_sum(float v) {
#pragma unroll
  for (int off = 16; off > 0; off >>= 1) v += __shfl_xor(v, off, 32);
  return v;
}

// ---------------------------------------------------------------------------
// f32 WMMA GEMM: C[M,N] = A[M,K] @ W[K,N] + bias*bias_scale  (opt. accum/relu)
// Each wave computes a 32x64 block: 2 M-tiles x 4 N-tiles of 16x16, using
// V_WMMA_F32_16X16X4_F32 with K stepped by 4.
// The 32-row A tile is SHARED by all 8 waves of the block: it is staged into
// LDS in 64-wide K chunks with gfx1250 async global->LDS copies
// (global_load_async_to_lds_b128, ASYNCcnt, s_wait_asynccnt), then WMMA A
// fragments are read from LDS (ds_load_b64). B streams from global (64B
// coalesced per 16 lanes, L2-resident across M-blocks).
// Block = 256 threads = 8 waves covering 512 columns; grid = (N/512, M/32).
// REQUIRES exact grid (N % 512 == 0, M % 32 == 0) - no partial-tile guard,
// so EXEC is all-1s for every WMMA and barriers are safe.
// ---------------------------------------------------------------------------
#define KC 64   // K-chunk staged in LDS

__global__ void gemm_bias_wmma(const float* __restrict__ A, const float* __restrict__ W,
                               const float* __restrict__ bias, float* __restrict__ C,
                               int M, int K, int N, float bias_scale, int accum, int relu) {
  __shared__ float Atile[32 * KC];              // 8 KB: rows m0..m0+31, KC K's

  const int lane = threadIdx.x & 31;
  const int wave = threadIdx.x >> 5;
  const int n0 = (blockIdx.x * 8 + wave) * 64;  // 64 cols per wave
  const int m0 = blockIdx.y * 32;               // 32 rows per block (shared A)

  const int half = lane >> 4;   // 0: K=0,1 lanes | 1: K=2,3 lanes
  const int l15  = lane & 15;
  const int akb  = half * 2;

  v8f acc[2][4];
#pragma unroll
  for (int i = 0; i < 2; ++i)
#pragma unroll
    for (int t = 0; t < 4; ++t)
      acc[i][t] = (v8f){0.f,0.f,0.f,0.f,0.f,0.f,0.f,0.f};

  // async-copy assignment: wave w copies elements [ (w*2+i)*128 , +128 )
  // of the 32x64 chunk; each lane moves 16B (4 floats).
  const int e0 = ((wave * 2 + 0) * 32 + lane) * 4;   // linear float index
  const int e1 = ((wave * 2 + 1) * 32 + lane) * 4;
  const int r0 = e0 / KC, kk0 = e0 % KC;
  const int r1 = e1 / KC, kk1 = e1 % KC;
  const unsigned loff0 = (unsigned)(size_t)(&Atile[e0]);
  const unsigned loff1 = (unsigned)(size_t)(&Atile[e1]);

  const float* As0 = &Atile[l15 * KC + akb];          // A frag rows, M-tile 0
  const float* As1 = &Atile[(16 + l15) * KC + akb];   // A frag rows, M-tile 1

  for (int k0 = 0; k0 < K; k0 += KC) {
    __syncthreads();  // all waves done reading previous chunk
    {
      const float* g0 = A + (size_t)(m0 + r0) * K + k0 + kk0;
      const float* g1 = A + (size_t)(m0 + r1) * K + k0 + kk1;
      asm volatile("global_load_async_to_lds_b128 %0, %1, off"
                   :: "v"(loff0), "v"(g0) : "memory");
      asm volatile("global_load_async_to_lds_b128 %0, %1, off"
                   :: "v"(loff1), "v"(g1) : "memory");
    }
    // prefetch next B chunk while waiting on the A copy
    if (k0 + KC < K)
      __builtin_prefetch(W + (size_t)(k0 + KC + akb) * N + n0 + l15, 0, 0);
    asm volatile("s_wait_asynccnt 0x0" ::: "memory");
    __syncthreads();  // LDS chunk visible to all waves

#pragma unroll 4
    for (int kk = 0; kk < KC; kk += 4) {
      v2f a0 = *(const v2f*)(As0 + kk);
      v2f a1 = *(const v2f*)(As1 + kk);
      const float* Wp = W + (size_t)(k0 + kk + akb) * N + n0 + l15;
#pragma unroll
      for (int t = 0; t < 4; ++t) {
        v2f b;
        b.x = Wp[t * 16];
        b.y = Wp[t * 16 + N];
        acc[0][t] = __builtin_amdgcn_wmma_f32_16x16x4_f32(false, a0, false, b, (short)0,
                                                          acc[0][t], false, false);
        acc[1][t] = __builtin_amdgcn_wmma_f32_16x16x4_f32(false, a1, false, b, (short)0,
                                                          acc[1][t], false, false);
      }
    }
  }

#pragma unroll
  for (int i = 0; i < 2; ++i) {
    const int rbase = m0 + i * 16 + half * 8;
#pragma unroll
    for (int t = 0; t < 4; ++t) {
      const int col = n0 + t * 16 + l15;
      const float bv = bias ? bias[col] * bias_scale : 0.f;
#pragma unroll
      for (int r = 0; r < 8; ++r) {
        float v = acc[i][t][r] + bv;
        float* cp = C + (size_t)(rbase + r) * N + col;
        if (accum) v += *cp;
        if (relu)  v = fmaxf(v, 0.f);
        *cp = v;
      }
    }
  }
}

// ---------------------------------------------------------------------------
// score[b,n] = sum_d tanh(p_feats[b,n,d] + att_h[b,d]) * w_alpha[d]
// (scalar bias to score is softmax-invariant, so omitted)
// One wave per (b,n) row.
// ---------------------------------------------------------------------------
__global__ void att_score_kernel(const float* __restrict__ p_feats, const float* __restrict__ att_h,
                                 const float* __restrict__ w_alpha, float* __restrict__ score,
                                 int BN, int L, int D) {
  const int widx = (int)((blockIdx.x * blockDim.x + threadIdx.x) >> 5);
  const int lane = threadIdx.x & 31;
  if (widx >= BN) return;
  const int b = widx / L, n = widx % L;
  const float* pf = p_feats + ((size_t)b * L + n) * D;
  const float* ah = att_h + (size_t)b * D;
  float acc = 0.f;
  for (int d = lane; d < D; d += 32) acc += tanhf(pf[d] + ah[d]) * w_alpha[d];
  acc = wave_sum(acc);
  if (lane == 0) score[widx] = acc;
}

// ---------------------------------------------------------------------------
// softmax over n (with optional mask -> MIN_VALUE), then
// att_res[b,d] (+)= sum_n w[b,n] * feats[b,n,d].  One block (256 thr) per b.
// ---------------------------------------------------------------------------
__global__ void softmax_wsum_kernel(const float* __restrict__ score, const float* __restrict__ feats,
                                    const int* __restrict__ mask, int mask_stride, int mask_off,
                                    float* __restrict__ att_res, int L, int D, int accum) {
  __shared__ float wgt[256];
  __shared__ float red[256];
  const int b = blockIdx.x, tid = threadIdx.x;

  float s = -INFINITY;
  if (tid < L) {
    s = score[b * L + tid];
    if (mask && mask[b * mask_stride + mask_off + tid]) s = MIN_VALUE;
  }
  red[tid] = s;
  __syncthreads();
  for (int o = 128; o > 0; o >>= 1) { if (tid < o) red[tid] = fmaxf(red[tid], red[tid + o]); __syncthreads(); }
  const float mx = red[0];
  __syncthreads();
  const float e = (tid < L) ? __expf(s - mx) : 0.f;
  red[tid] = e;
  __syncthreads();
  for (int o = 128; o > 0; o >>= 1) { if (tid < o) red[tid] += red[tid + o]; __syncthreads(); }
  const float inv = 1.f / red[0];
  __syncthreads();
  wgt[tid] = e * inv;
  __syncthreads();

  const float* fb = feats + (size_t)b * L * D;
  for (int d = tid; d < D; d += 256) {
    float acc = 0.f;
    for (int n = 0; n < L; ++n) acc += wgt[n] * fb[(size_t)n * D + d];
    float* o = att_res + (size_t)b * D + d;
    if (accum) *o += acc; else *o = acc;
  }
}

// ---------------------------------------------------------------------------
// LSTM gates / cell update / fake_box. One thread per (b, j).
// sums[b, 0:6R]; a2c[b, 0:2R] already includes both attention terms + 2*bias.
// ---------------------------------------------------------------------------
__global__ void lstm_kernel(const float* __restrict__ sums, const float* __restrict__ a2c,
                            const float* __restrict__ c_prev, float* __restrict__ out_output,
                            float* __restrict__ out_next_h, float* __restrict__ out_next_c,
                            float* __restrict__ fake_box) {
  const int idx = blockIdx.x * blockDim.x + threadIdx.x;
  if (idx >= BSZ * RNN) return;
  const int b = idx / RNN, j = idx % RNN;
  const float* sr = sums + (size_t)b * 6 * RNN;
  const float ig = 1.f / (1.f + __expf(-sr[j]));
  const float fg = 1.f / (1.f + __expf(-sr[RNN + j]));
  const float og = 1.f / (1.f + __expf(-sr[2 * RNN + j]));
  const float sg = 1.f / (1.f + __expf(-sr[3 * RNN + j]));
  const float* ar = a2c + (size_t)b * 2 * RNN;
  const float t1 = sr[4 * RNN + j] + ar[j];
  const float t2 = sr[5 * RNN + j] + ar[RNN + j];
  const float itr = fmaxf(t1, t2);
  const float c  = fg * c_prev[idx] + ig * itr;
  const float th = tanhf(c);
  const float h  = og * th;
  out_output[idx] = h;
  out_next_h[idx] = h;
  out_next_c[idx] = c;
  fake_box[idx]   = sg * th;
}

// ---------------------------------------------------------------------------
// det_prob[b,j] = sum_d tanh(img_all[b,j,d] + h_embed[b,d]) * W_alphap[d] + b_alphap
// with img_all[b,0]=fr_embed[b], img_all[b,1+n]=p_pool_feats[b,n]; mask -> MIN.
// One wave per (b,j).
// ---------------------------------------------------------------------------
__global__ void det_prob_kernel(const float* __restrict__ fr_embed, const float* __restrict__ p_pool_feats,
                                const float* __restrict__ h_embed, const float* __restrict__ w_alphap,
                                const float* __restrict__ b_alphap, const int* __restrict__ pnt_mask,
                                float* __restrict__ det_out) {
  const int J = RSZ + 1;
  const int widx = (int)((blockIdx.x * blockDim.x + threadIdx.x) >> 5);
  const int lane = threadIdx.x & 31;
  if (widx >= BSZ * J) return;
  const int b = widx / J, j = widx % J;
  const float* base = (j == 0) ? (fr_embed + (size_t)b * ATTH)
                               : (p_pool_feats + ((size_t)b * RSZ + (j - 1)) * ATTH);
  const float* he = h_embed + (size_t)b * ATTH;
  float acc = 0.f;
  for (int d = lane; d < ATTH; d += 32) acc += tanhf(base[d] + he[d]) * w_alphap[d];
  acc = wave_sum(acc) + b_alphap[0];
  if (pnt_mask[b * J + j]) acc = MIN_VALUE;
  if (lane == 0) det_out[widx] = acc;
}

// ---------------------------------------------------------------------------
extern "C" void kernel_launch(void* const* d_in, const int* in_sizes, int n_in,
                              void* d_out, int out_size, void* d_ws, size_t ws_size,
                              hipStream_t stream) {
  const float* xt           = (const float*)d_in[0];
  const float* att_feats    = (const float*)d_in[2];
  const float* p_att_feats  = (const float*)d_in[3];
  const float* pool_feats   = (const float*)d_in[4];
  const float* p_pool_feats = (const float*)d_in[5];
  const int*   att_mask     = (const int*)d_in[6];
  const int*   pnt_mask     = (const int*)d_in[7];
  const float* h_prev       = (const float*)d_in[8];   // (1,B,RNN) flat
  const float* c_prev       = (const float*)d_in[9];
  const float* W_h2att = (const float*)d_in[10]; const float* b_h2att = (const float*)d_in[11];
  const float* W_alpha1 = (const float*)d_in[12];                       // b_alpha1 (13): softmax-invariant
  const float* W_h1att = (const float*)d_in[14]; const float* b_h1att = (const float*)d_in[15];
  const float* W_alpha2 = (const float*)d_in[16];                       // b_alpha2 (17): softmax-invariant
  const float* W_i2h  = (const float*)d_in[18]; const float* b_i2h  = (const float*)d_in[19];
  const float* W_h2h  = (const float*)d_in[20]; const float* b_h2h  = (const float*)d_in[21];
  const float* W_a2c1 = (const float*)d_in[22]; const float* b_a2c1 = (const float*)d_in[23];
  const float* W_ffc1 = (const float*)d_in[24]; const float* b_ffc1 = (const float*)d_in[25];
  const float* W_ffc2 = (const float*)d_in[26]; const float* b_ffc2 = (const float*)d_in[27];
  const float* W_hfc1 = (const float*)d_in[28]; const float* b_hfc1 = (const float*)d_in[29];
  const float* W_alphap = (const float*)d_in[30]; const float* b_alphap = (const float*)d_in[31];

  // outputs: output | det_prob | next_h | next_c
  float* out_output = (float*)d_out;
  float* out_det    = out_output + BSZ * RNN;
  float* out_next_h = out_det + BSZ * (RSZ + 1);
  float* out_next_c = out_next_h + BSZ * RNN;

  // workspace layout (floats)
  float* ws       = (float*)d_ws;
  float* att_h1   = ws;                         // 256*512
  float* att_h2   = att_h1 + BSZ * ATTH;        // 256*512
  float* score1   = att_h2 + BSZ * ATTH;        // 256*196
  float* score2   = score1 + BSZ * LSZ;         // 256*100
  float* att_res  = score2 + BSZ * RSZ;         // 256*1024 (att1 + att2)
  float* sums     = att_res + BSZ * RNN;        // 256*6144
  float* a2c      = sums + BSZ * 6 * RNN;       // 256*2048
  float* fake_box = a2c + BSZ * 2 * RNN;        // 256*1024
  float* fake_reg = fake_box + BSZ * RNN;       // 256*1024
  float* fr_embed = fake_reg + BSZ * RNN;       // 256*512
  float* h_embed  = fr_embed + BSZ * ATTH;      // 256*512

  const dim3 blk(256);

  // attention projections: att_h = h_prev @ W + b   [256,1024,512]
  gemm_bias_wmma<<<dim3(1, 8), blk, 0, stream>>>(h_prev, W_h2att, b_h2att, att_h1,
                                                 BSZ, RNN, ATTH, 1.f, 0, 0);
  gemm_bias_wmma<<<dim3(1, 8), blk, 0, stream>>>(h_prev, W_h1att, b_h1att, att_h2,
                                                 BSZ, RNN, ATTH, 1.f, 0, 0);

  // scores (wave per row)
  att_score_kernel<<<(BSZ * LSZ) / 8, blk, 0, stream>>>(p_att_feats, att_h1, W_alpha1,
                                                        score1, BSZ * LSZ, LSZ, ATTH);
  att_score_kernel<<<(BSZ * RSZ) / 8, blk, 0, stream>>>(p_pool_feats, att_h2, W_alpha2,
                                                        score2, BSZ * RSZ, RSZ, ATTH);

  // softmax + weighted feature sum: att_res = att1; att_res += att2 (masked)
  softmax_wsum_kernel<<<BSZ, blk, 0, stream>>>(score1, att_feats, nullptr, 0, 0,
                                               att_res, LSZ, RNN, 0);
  softmax_wsum_kernel<<<BSZ, blk, 0, stream>>>(score2, pool_feats, att_mask, RSZ + 1, 1,
                                               att_res, RSZ, RNN, 1);

  // sums = xt @ W_i2h + b_i2h + h_prev @ W_h2h + b_h2h   [256,1024,6144]
  gemm_bias_wmma<<<dim3(12, 8), blk, 0, stream>>>(xt, W_i2h, b_i2h, sums,
                                                  BSZ, ENCSZ, 6 * RNN, 1.f, 0, 0);
  gemm_bias_wmma<<<dim3(12, 8), blk, 0, stream>>>(h_prev, W_h2h, b_h2h, sums,
                                                  BSZ, RNN, 6 * RNN, 1.f, 1, 0);

  // a2c = (att1+att2) @ W_a2c1 + 2*b_a2c1   [256,1024,2048]
  gemm_bias_wmma<<<dim3(4, 8), blk, 0, stream>>>(att_res, W_a2c1, b_a2c1, a2c,
                                                 BSZ, RNN, 2 * RNN, 2.f, 0, 0);

  // LSTM elementwise (writes output, next_h, next_c, fake_box)
  lstm_kernel<<<(BSZ * RNN) / 256, blk, 0, stream>>>(sums, a2c, c_prev,
                                                     out_output, out_next_h, out_next_c, fake_box);

  // fake_region = relu(fake_box @ W_ffc1 + b)   [256,1024,1024]
  gemm_bias_wmma<<<dim3(2, 8), blk, 0, stream>>>(fake_box, W_ffc1, b_ffc1, fake_reg,
                                                 BSZ, RNN, RNN, 1.f, 0, 1);
  // fr_embed = fake_region @ W_ffc2 + b   [256,1024,512]
  gemm_bias_wmma<<<dim3(1, 8), blk, 0, stream>>>(fake_reg, W_ffc2, b_ffc2, fr_embed,
                                                 BSZ, RNN, ATTH, 1.f, 0, 0);
  // h_embed = output @ W_hfc1 + b   [256,1024,512]
  gemm_bias_wmma<<<dim3(1, 8), blk, 0, stream>>>(out_output, W_hfc1, b_hfc1, h_embed,
                                                 BSZ, RNN, ATTH, 1.f, 0, 0);

  // det_prob (wave per (b,j))
  det_prob_kernel<<<(BSZ * (RSZ + 1)) / 8, blk, 0, stream>>>(fr_embed, p_pool_feats, h_embed,
                                                             W_alphap, b_alphap, pnt_mask, out_det);
}